// UGNet_MoE_17961553232379
// MI455X (gfx1250) — compile-verified
//
#include <hip/hip_runtime.h>
#include <math.h>

typedef float v2f __attribute__((ext_vector_type(2)));
typedef float v8f __attribute__((ext_vector_type(8)));

#define BSZ     256
#define TSEQ    60
#define INDIM   520
#define GATE    8
#define MOE_RAW 512
#define MOE_IN  992   // 512 + 8*60
#define HID     752   // 512 + 240
#define OUTD    512
#define NE      8

// ---------------------------------------------------------------------------
// Kernel 1: assemble Xm[b] = concat(X[b, T-1, 0:512], X[b, :, 512:520].flat)
// ---------------------------------------------------------------------------
__global__ void k_build_xm(const float* __restrict__ X, float* __restrict__ Xm) {
  int idx = blockIdx.x * blockDim.x + threadIdx.x;
  const int total = BSZ * MOE_IN;
  if (idx >= total) return;
  int b = idx / MOE_IN;
  int c = idx - b * MOE_IN;
  float v;
  if (c < MOE_RAW) {
    v = X[((size_t)b * TSEQ + (TSEQ - 1)) * INDIM + c];
  } else {
    int q = c - MOE_RAW;
    int t = q >> 3;
    int j = q & 7;
    v = X[((size_t)b * TSEQ + t) * INDIM + MOE_RAW + j];
  }
  Xm[idx] = v;
}

// ---------------------------------------------------------------------------
// Kernel 2: 2-layer GRU (H=8) + softmax gate. One wave32 per batch element.
// Lane g (g<24) holds gate-row g of Wih/Whh in registers. Recurrent state h
// is kept replicated per-lane and rebuilt each step via __shfl (wave32).
// ---------------------------------------------------------------------------
__device__ __forceinline__ float sigm(float x) { return 1.0f / (1.0f + __expf(-x)); }

__global__ void k_gru(const float* __restrict__ X,
                      const float* __restrict__ Wih0, const float* __restrict__ Whh0,
                      const float* __restrict__ bih0, const float* __restrict__ bhh0,
                      const float* __restrict__ Wih1, const float* __restrict__ Whh1,
                      const float* __restrict__ bih1, const float* __restrict__ bhh1,
                      float* __restrict__ Omega) {
  const int b = blockIdx.x;          // 256 blocks
  const int g = threadIdx.x;         // 32 lanes, rows g<24 meaningful
  const int k = g & 7;

  float wih0[8], whh0[8], wih1[8], whh1[8];
  float bi0 = 0.f, bh0 = 0.f, bi1 = 0.f, bh1 = 0.f;
  if (g < 24) {
    #pragma unroll
    for (int j = 0; j < 8; j++) {
      wih0[j] = Wih0[g * 8 + j];  whh0[j] = Whh0[g * 8 + j];
      wih1[j] = Wih1[g * 8 + j];  whh1[j] = Whh1[g * 8 + j];
    }
    bi0 = bih0[g]; bh0 = bhh0[g]; bi1 = bih1[g]; bh1 = bhh1[g];
  } else {
    #pragma unroll
    for (int j = 0; j < 8; j++) { wih0[j]=0.f; whh0[j]=0.f; wih1[j]=0.f; whh1[j]=0.f; }
  }

  float h0[8], h1[8];
  #pragma unroll
  for (int j = 0; j < 8; j++) { h0[j] = 0.f; h1[j] = 0.f; }

  for (int t = 0; t < TSEQ; t++) {
    const float* xp = X + ((size_t)b * TSEQ + t) * INDIM + MOE_RAW;
    float xt[8];
    #pragma unroll
    for (int j = 0; j < 8; j++) xt[j] = xp[j];   // same addr all lanes -> bcast

    // ---- layer 0 ----
    {
      float gi = bi0, gh = bh0;
      #pragma unroll
      for (int j = 0; j < 8; j++) { gi += wih0[j] * xt[j]; gh += whh0[j] * h0[j]; }
      float s  = gi + gh;
      float rk = sigm(__shfl(s,  k, 32));
      float zk = sigm(__shfl(s,  8 + k, 32));
      float ik = __shfl(gi, 16 + k, 32);
      float hk = __shfl(gh, 16 + k, 32);
      float nk = tanhf(ik + rk * hk);
      float hnew = (1.f - zk) * nk + zk * h0[k];
      #pragma unroll
      for (int j = 0; j < 8; j++) h0[j] = __shfl(hnew, j, 32);
    }
    // ---- layer 1 (input = h0 just updated) ----
    {
      float gi = bi1, gh = bh1;
      #pragma unroll
      for (int j = 0; j < 8; j++) { gi += wih1[j] * h0[j]; gh += whh1[j] * h1[j]; }
      float s  = gi + gh;
      float rk = sigm(__shfl(s,  k, 32));
      float zk = sigm(__shfl(s,  8 + k, 32));
      float ik = __shfl(gi, 16 + k, 32);
      float hk = __shfl(gh, 16 + k, 32);
      float nk = tanhf(ik + rk * hk);
      float hnew = (1.f - zk) * nk + zk * h1[k];
      #pragma unroll
      for (int j = 0; j < 8; j++) h1[j] = __shfl(hnew, j, 32);
    }
  }

  // softmax over h1 (replicated in every lane); lanes 0..7 write
  float m = h1[0];
  #pragma unroll
  for (int j = 1; j < 8; j++) m = fmaxf(m, h1[j]);
  float s = 0.f, ex[8];
  #pragma unroll
  for (int j = 0; j < 8; j++) { ex[j] = __expf(h1[j] - m); s += ex[j]; }
  if (g < 8) Omega[(size_t)b * NE + g] = ex[g] / s;
}

// ---------------------------------------------------------------------------
// Kernel 3: blended expert GEMM. One wave per 16x16 output tile.
//   y[b,h] = act( sum_e Om[b,e] * ( x[b,:] @ w[e,:,h] + bias[e,h] ) )
// I, H compile-time so all B-matrix addressing splits into:
//   - wave-uniform SGPR part: e*I*H + k0*H (strength-reduced s_add per iter)
//   - loop-invariant per-lane VGPR offset: hn0 + l16 + (2*half)*H
// keeping inner-loop VALU near zero (f32 WMMA does not co-execute with VALU).
// A frag (16x4 f32, 2 VGPR): lanes 0-15 -> K0/K1 of row M=lane,
//                            lanes 16-31 -> K2/K3 of row M=lane-16.
// B frag (4x16 f32, 2 VGPR): lanes 0-15 -> N=lane of rows K0/K1,
//                            lanes 16-31 -> N=lane-16 of rows K2/K3.
// C/D (16x16 f32, 8 VGPR): vgpr r, lane l -> M = r + 8*(l>>4), N = l&15.
// ---------------------------------------------------------------------------
template <int I, int H, bool ACT>
__global__ void k_blend(const float* __restrict__ x,
                        const float* __restrict__ w,     // [NE, I, H]
                        const float* __restrict__ bias,  // [NE, H]
                        const float* __restrict__ Om,    // [BSZ, NE]
                        float* __restrict__ y) {         // [BSZ, H]
  const int lane = threadIdx.x;        // 32
  const int half = lane >> 4;          // 0|1
  const int l16  = lane & 15;
  const int hn0  = blockIdx.x * 16;    // output col tile
  const int bt0  = blockIdx.y * 16;    // batch row tile

  v8f acc[NE];
  const v8f vzero = {0.f, 0.f, 0.f, 0.f, 0.f, 0.f, 0.f, 0.f};
  #pragma unroll
  for (int e = 0; e < NE; e++) acc[e] = vzero;

  const float* xrow = x + (size_t)(bt0 + l16) * I + 2 * half;  // A stream
  const int lane_off = hn0 + l16 + 2 * half * H;               // per-lane, invariant

  #pragma unroll 2
  for (int k0 = 0; k0 < I; k0 += 4) {
    v2f a = *reinterpret_cast<const v2f*>(xrow + k0);          // 8B aligned
    const size_t ko = (size_t)k0 * H;                          // uniform
    #pragma unroll
    for (int e = 0; e < NE; e++) {
      const size_t uni = (size_t)e * I * H + ko;               // uniform (SGPR)
      v2f bb;
      bb.x = w[uni + lane_off];
      bb.y = w[uni + H + lane_off];
      acc[e] = __builtin_amdgcn_wmma_f32_16x16x4_f32(
          /*neg_a=*/false, a, /*neg_b=*/false, bb,
          /*c_mod=*/(short)0, acc[e], /*reuse_a=*/false, /*reuse_b=*/false);
    }
  }

  // blended bias for this lane's column
  float bcol[NE];
  #pragma unroll
  for (int e = 0; e < NE; e++) bcol[e] = bias[(size_t)e * H + hn0 + l16];

  #pragma unroll
  for (int r = 0; r < 8; r++) {
    const int row = bt0 + r + 8 * half;
    const float* om = Om + (size_t)row * NE;
    float v = 0.f;
    #pragma unroll
    for (int e = 0; e < NE; e++) v += om[e] * (acc[e][r] + bcol[e]);
    if (ACT) v = (v > 0.f) ? v : (__expf(v) - 1.f);            // ELU, alpha=1
    y[(size_t)row * H + hn0 + l16] = v;
  }
}

// ---------------------------------------------------------------------------
// Launch
// ---------------------------------------------------------------------------
extern "C" void kernel_launch(void* const* d_in, const int* in_sizes, int n_in,
                              void* d_out, int out_size, void* d_ws, size_t ws_size,
                              hipStream_t stream) {
  const float* X    = (const float*)d_in[0];
  const float* Wih0 = (const float*)d_in[1];
  const float* Whh0 = (const float*)d_in[2];
  const float* bih0 = (const float*)d_in[3];
  const float* bhh0 = (const float*)d_in[4];
  const float* Wih1 = (const float*)d_in[5];
  const float* Whh1 = (const float*)d_in[6];
  const float* bih1 = (const float*)d_in[7];
  const float* bhh1 = (const float*)d_in[8];
  const float* w1   = (const float*)d_in[9];
  const float* b1   = (const float*)d_in[10];
  const float* w2   = (const float*)d_in[11];
  const float* b2   = (const float*)d_in[12];
  const float* w3   = (const float*)d_in[13];
  const float* b3   = (const float*)d_in[14];

  char* ws = (char*)d_ws;
  float* Xm    = (float*)(ws);                       // 256*992*4  = 1,015,808
  float* Omega = (float*)(ws + 1015808);             // 256*8*4    = 8,192
  float* a1    = (float*)(ws + 1024000);             // 256*752*4  = 770,048
  float* a2    = (float*)(ws + 1794048);             // 256*752*4  = 770,048
  float* Y     = (float*)d_out;                      // 256*512

  // 1) gather Xm
  {
    int total = BSZ * MOE_IN;
    k_build_xm<<<(total + 255) / 256, 256, 0, stream>>>(X, Xm);
  }
  // 2) GRU + softmax gate: one wave per batch element
  k_gru<<<BSZ, 32, 0, stream>>>(X, Wih0, Whh0, bih0, bhh0,
                                Wih1, Whh1, bih1, bhh1, Omega);
  // 3) blended expert layers (fp32 WMMA)
  k_blend<MOE_IN, HID,  true ><<<dim3(HID / 16,  BSZ / 16), 32, 0, stream>>>(Xm, w1, b1, Omega, a1);
  k_blend<HID,    HID,  true ><<<dim3(HID / 16,  BSZ / 16), 32, 0, stream>>>(a1, w2, b2, Omega, a2);
  k_blend<HID,    OUTD, false><<<dim3(OUTD / 16, BSZ / 16), 32, 0, stream>>>(a2, w3, b3, Omega, Y);
}